// LinkPredictionModel_18391049961797
// MI455X (gfx1250) — compile-verified
//
#include <hip/hip_runtime.h>

typedef __attribute__((ext_vector_type(2))) float v2f;
typedef __attribute__((ext_vector_type(8))) float v8f;

#define NNODES 50000
#define NEDGES 1000000
#define DFEAT  64      // node feature width (din = dh = dout = 64)
#define DEDGE  16
#define KPACK  144     // 64 (self) + 64 (neighbor mean) + 16 (edge mean)
#define MTILES (NNODES / 16)   // 3125, exact

// ---------------- elementwise / setup kernels ----------------

__global__ void zero_f32_kernel(float* p, long n) {
    long i = (long)blockIdx.x * blockDim.x + threadIdx.x;
    if (i < n) p[i] = 0.0f;
}

__global__ void init_cnt_kernel(float* cnt, int n) {
    int i = blockIdx.x * blockDim.x + threadIdx.x;
    if (i < n) cnt[i] = 1.0f;   // self-loop contributes 1 to the degree
}

// Per-edge: accumulate edge_attr into Eagg[dst] and bump cnt[dst]. Done ONCE
// (edge_attr and degrees are identical for both layers).
__global__ void edge_attr_scatter_kernel(const int* __restrict__ dst,
                                         const float* __restrict__ ea,
                                         float* __restrict__ eagg,
                                         float* __restrict__ cnt) {
    long t = (long)blockIdx.x * blockDim.x + threadIdx.x;
    if (t >= (long)NEDGES * DEDGE) return;
    int e = (int)(t >> 4);
    int j = (int)(t & 15);
    int d = dst[e];
    atomicAdd(&eagg[(long)d * DEDGE + j], ea[(long)e * DEDGE + j]);
    if (j == 0) atomicAdd(&cnt[d], 1.0f);
}

// Per-edge: Xagg[dst] += xin[src] (64 floats). One thread per (edge, feature).
__global__ void edge_feat_scatter_kernel(const int* __restrict__ src,
                                         const int* __restrict__ dst,
                                         const float* __restrict__ xin,
                                         float* __restrict__ xagg) {
    long t = (long)blockIdx.x * blockDim.x + threadIdx.x;
    if (t >= (long)NEDGES * DFEAT) return;
    int e = (int)(t >> 6);
    int f = (int)(t & 63);
    int s = src[e];
    int d = dst[e];
    atomicAdd(&xagg[(long)d * DFEAT + f], xin[(long)s * DFEAT + f]);
}

// Build packed per-node feature row F[i] = [ x_i | (Xagg_i + x_i)/cnt_i | Eagg_i/cnt_i ]
__global__ void build_feat_kernel(const float* __restrict__ xin,
                                  const float* __restrict__ xagg,
                                  const float* __restrict__ eagg,
                                  const float* __restrict__ cnt,
                                  float* __restrict__ F) {
    int i = blockIdx.x;
    int f = threadIdx.x;
    if (f >= KPACK) return;
    float rc = 1.0f / cnt[i];
    float v;
    if (f < 64) {
        v = xin[(long)i * DFEAT + f];
    } else if (f < 128) {
        int g = f - 64;
        v = (xagg[(long)i * DFEAT + g] + xin[(long)i * DFEAT + g]) * rc;
    } else {
        v = eagg[(long)i * DEDGE + (f - 128)] * rc;
    }
    F[(long)i * KPACK + f] = v;
}

// Pack [Ws (64x64); Wn (80x64)] into a SWIZZLED 144x64 weight block:
//   Wp[(k>>1)*128 + col*2 + (k&1)] = W[k][col]
// so that one lane's B-operand pair for a K-pair is 8 contiguous, 8B-aligned
// bytes -> single ds_load_b64 into an even VGPR pair (no repacking movs).
__global__ void pack_weights_kernel(const float* __restrict__ Ws,
                                    const float* __restrict__ Wn,
                                    float* __restrict__ Wp) {
    int r = blockIdx.x;     // k: 0..143
    int c = threadIdx.x;    // col: 0..63
    float v = (r < 64) ? Ws[r * 64 + c] : Wn[(r - 64) * 64 + c];
    Wp[(r >> 1) * 128 + c * 2 + (r & 1)] = v;
}

__global__ void pack_bias_kernel(const float* __restrict__ bs,
                                 const float* __restrict__ bn,
                                 float* __restrict__ bp) {
    int c = threadIdx.x;
    if (c < 64) bp[c] = bs[c] + bn[c];
}

// ---------------- WMMA GEMM: out = relu(F[N,144] @ W[144,64] + bias) ----------------
// Block = 128 threads = 4 waves; each wave owns one 16-row M tile and computes
// the full 16x64 output via 4 f32 accumulators, K-loop of 36 steps of 4.
// Weights live in LDS in the swizzled [kpair][col][2] layout.
__global__ void gemm_relu_kernel(const float* __restrict__ F,
                                 const float* __restrict__ Wp,
                                 const float* __restrict__ bias,
                                 float* __restrict__ out) {
    __shared__ __align__(16) float sW[KPACK * 64];   // swizzled: [72][64][2]
    __shared__ __align__(16) float sB[64];

    for (int idx = threadIdx.x; idx < KPACK * 64; idx += blockDim.x)
        sW[idx] = Wp[idx];                            // straight coalesced copy
    if (threadIdx.x < 64) sB[threadIdx.x] = bias[threadIdx.x];
    __syncthreads();

    const int wave = threadIdx.x >> 5;
    const int lane = threadIdx.x & 31;
    const int tile = blockIdx.x * 4 + wave;
    if (tile >= MTILES) return;          // wave-uniform; surviving waves keep EXEC all-1s

    const int m0   = tile * 16;
    const int half = lane >> 4;          // 0: K pair k0..k0+1, 1: K pair k0+2..k0+3
    const int sub  = lane & 15;          // row (for A) / column (for B,C,D)

    const float* __restrict__ Arow = F + (long)(m0 + sub) * KPACK;

    v8f acc[4] = {};

    for (int k0 = 0; k0 < KPACK; k0 += 4) {
        const int kp = (k0 >> 1) + half;               // K-pair index
        v2f a = *(const v2f*)(Arow + k0 + half * 2);   // 8B-aligned global_load_b64
#pragma unroll
        for (int nt = 0; nt < 4; ++nt) {
            v2f b = *(const v2f*)(sW + kp * 128 + (nt * 16 + sub) * 2);  // ds_load_b64
            acc[nt] = __builtin_amdgcn_wmma_f32_16x16x4_f32(
                /*neg_a=*/false, a, /*neg_b=*/false, b,
                /*c_mod=*/(short)0, acc[nt],
                /*reuse_a=*/false, /*reuse_b=*/false);
        }
    }

#pragma unroll
    for (int nt = 0; nt < 4; ++nt) {
#pragma unroll
        for (int r = 0; r < 8; ++r) {
            int row = m0 + r + half * 8;
            int col = nt * 16 + sub;
            float v = acc[nt][r] + sB[col];
            out[(long)row * DFEAT + col] = v > 0.0f ? v : 0.0f;
        }
    }
}

// ---------------- driver ----------------

extern "C" void kernel_launch(void* const* d_in, const int* in_sizes, int n_in,
                              void* d_out, int out_size, void* d_ws, size_t ws_size,
                              hipStream_t stream) {
    (void)in_sizes; (void)n_in; (void)out_size; (void)ws_size;

    const float* x    = (const float*)d_in[0];
    const int*   ei   = (const int*)d_in[1];     // [2, E]: row0 = src, row1 = dst
    const float* ea   = (const float*)d_in[2];
    const float* W1n  = (const float*)d_in[3];
    const float* b1n  = (const float*)d_in[4];
    const float* W1s  = (const float*)d_in[5];
    const float* b1s  = (const float*)d_in[6];
    const float* W2n  = (const float*)d_in[7];
    const float* b2n  = (const float*)d_in[8];
    const float* W2s  = (const float*)d_in[9];
    const float* b2s  = (const float*)d_in[10];
    float* out = (float*)d_out;

    const int* srcp = ei;
    const int* dstp = ei + NEDGES;

    // workspace layout (floats)
    float* ws   = (float*)d_ws;
    float* Xagg = ws;                                 // N*64
    float* Eagg = Xagg + (long)NNODES * DFEAT;        // N*16
    float* cnt  = Eagg + (long)NNODES * DEDGE;        // N
    float* F    = cnt  + NNODES;                      // N*144
    float* h    = F    + (long)NNODES * KPACK;        // N*64
    float* Wp1  = h    + (long)NNODES * DFEAT;        // 144*64 (swizzled)
    float* Wp2  = Wp1  + KPACK * 64;                  // 144*64 (swizzled)
    float* bp1  = Wp2  + KPACK * 64;                  // 64
    float* bp2  = bp1  + 64;                          // 64

    const int TB = 256;
    const long nEagg = (long)NNODES * DEDGE;
    const long nXagg = (long)NNODES * DFEAT;

    // --- once: degrees + edge_attr aggregation + packed weights/biases ---
    zero_f32_kernel<<<(int)((nEagg + TB - 1) / TB), TB, 0, stream>>>(Eagg, nEagg);
    init_cnt_kernel<<<(NNODES + TB - 1) / TB, TB, 0, stream>>>(cnt, NNODES);
    edge_attr_scatter_kernel<<<(int)(((long)NEDGES * DEDGE + TB - 1) / TB), TB, 0, stream>>>(
        dstp, ea, Eagg, cnt);
    pack_weights_kernel<<<KPACK, 64, 0, stream>>>(W1s, W1n, Wp1);
    pack_weights_kernel<<<KPACK, 64, 0, stream>>>(W2s, W2n, Wp2);
    pack_bias_kernel<<<1, 64, 0, stream>>>(b1s, b1n, bp1);
    pack_bias_kernel<<<1, 64, 0, stream>>>(b2s, b2n, bp2);

    const int gemmBlocks = (MTILES + 3) / 4;

    // --- layer 1: x -> h ---
    zero_f32_kernel<<<(int)((nXagg + TB - 1) / TB), TB, 0, stream>>>(Xagg, nXagg);
    edge_feat_scatter_kernel<<<(int)(((long)NEDGES * DFEAT + TB - 1) / TB), TB, 0, stream>>>(
        srcp, dstp, x, Xagg);
    build_feat_kernel<<<NNODES, 160, 0, stream>>>(x, Xagg, Eagg, cnt, F);
    gemm_relu_kernel<<<gemmBlocks, 128, 0, stream>>>(F, Wp1, bp1, h);

    // --- layer 2: h -> out ---
    zero_f32_kernel<<<(int)((nXagg + TB - 1) / TB), TB, 0, stream>>>(Xagg, nXagg);
    edge_feat_scatter_kernel<<<(int)(((long)NEDGES * DFEAT + TB - 1) / TB), TB, 0, stream>>>(
        srcp, dstp, h, Xagg);
    build_feat_kernel<<<NNODES, 160, 0, stream>>>(h, Xagg, Eagg, cnt, F);
    gemm_relu_kernel<<<gemmBlocks, 128, 0, stream>>>(F, Wp2, bp2, out);
}